// InFoRM_GNN_90374701843050
// MI455X (gfx1250) — compile-verified
//
#include <hip/hip_runtime.h>

typedef float v2f __attribute__((ext_vector_type(2)));
typedef float v8f __attribute__((ext_vector_type(8)));

#define N_NODES 100000
#define N_EDGES 1600000
#define NFEAT   256
#define NHID    32
#define NCLASS  8

#define TILE_ROWS 128
#define CHUNK_K   64
#define XPITCH    68   // floats per LDS row (64 + 4 pad -> conflict-free ds_load_b64)

// ---------------------------------------------------------------------------
// Kernel 0: Wc = W1 @ Wfc (256x32 @ 32x8 -> 256x8), bc = b1 @ Wfc + bfc (8)
// ---------------------------------------------------------------------------
__global__ __launch_bounds__(256) void precompute_kernel(
    const float* __restrict__ W1, const float* __restrict__ b1,
    const float* __restrict__ Wfc, const float* __restrict__ bfc,
    float* __restrict__ Wc, float* __restrict__ bc) {
  __shared__ float sWfc[NHID * NCLASS];
  int t = threadIdx.x;
  if (t < NHID * NCLASS) sWfc[t] = Wfc[t];
  __syncthreads();

  float acc[NCLASS];
#pragma unroll
  for (int j = 0; j < NCLASS; ++j) acc[j] = 0.f;
  const float* w1row = W1 + t * NHID;
#pragma unroll
  for (int m = 0; m < NHID; ++m) {
    float a = w1row[m];
#pragma unroll
    for (int j = 0; j < NCLASS; ++j) acc[j] += a * sWfc[m * NCLASS + j];
  }
#pragma unroll
  for (int j = 0; j < NCLASS; ++j) Wc[t * NCLASS + j] = acc[j];

  if (t < NCLASS) {
    float s = bfc[t];
    for (int m = 0; m < NHID; ++m) s += b1[m] * sWfc[m * NCLASS + t];
    bc[t] = s;
  }
}

// ---------------------------------------------------------------------------
// Degree pipeline: init to 1 (self-loop), scatter-count, rsqrt in place.
// ---------------------------------------------------------------------------
__global__ __launch_bounds__(256) void deg_init_kernel(float* __restrict__ deg) {
  int i = blockIdx.x * blockDim.x + threadIdx.x;
  if (i < N_NODES) deg[i] = 1.0f;
}

__global__ __launch_bounds__(256) void deg_edge_kernel(
    const int* __restrict__ dst, float* __restrict__ deg) {
  int e = blockIdx.x * blockDim.x + threadIdx.x;
  if (e < N_EDGES) unsafeAtomicAdd(&deg[dst[e]], 1.0f);
}

__global__ __launch_bounds__(256) void dinv_kernel(float* __restrict__ deg) {
  int i = blockIdx.x * blockDim.x + threadIdx.x;
  if (i < N_NODES) deg[i] = __frsqrt_rn(deg[i]);
}

// ---------------------------------------------------------------------------
// hc = x @ Wc via V_WMMA_F32_16X16X4_F32, with x tiles staged through LDS by
// async global->LDS copies (ASYNCcnt), double buffered across 4 K-chunks.
// Block: 256 threads = 8 waves; wave w owns rows 16w..16w+15 of a 128-row tile.
// ---------------------------------------------------------------------------
#define WAIT_ASYNC(n) asm volatile("s_wait_asynccnt " #n ::: "memory")

__global__ __launch_bounds__(256) void gemm_wmma_kernel(
    const float* __restrict__ x, const float* __restrict__ Wc,
    float* __restrict__ hc) {
  __shared__ float lds_w[NFEAT * 16];           // Wc padded N=8->16, 16 KB
  __shared__ float xs[2][TILE_ROWS * XPITCH];   // double-buffered x tile, 69.6 KB

  int t = threadIdx.x;
  for (int i = t; i < NFEAT * 16; i += 256) {
    int k = i >> 4, n = i & 15;
    lds_w[i] = (n < NCLASS) ? Wc[k * NCLASS + n] : 0.f;
  }

  const int row0 = blockIdx.x * TILE_ROWS;

  // Staging map: thread t copies 16B per op, 8 ops per 128x64 chunk.
  // Op i: local row = (t>>4) + 16*i, cols = (t&15)*4 .. +3  (wave = 2 rows x 256B)
  const int srow = t >> 4;
  const int scol = (t & 15) << 2;
  const unsigned long long xbase = (unsigned long long)x;

  auto issue = [&](int c) {
    const int k0g = c * CHUNK_K;
    float* buf = &xs[c & 1][0];
#pragma unroll
    for (int i = 0; i < 8; ++i) {
      int r = srow + (i << 4);
      int grow = row0 + r;
      if (grow >= N_NODES) grow = N_NODES - 1;  // clamp tail rows
      unsigned goff  = (unsigned)((grow * NFEAT + k0g + scol) << 2);
      unsigned laddr = (unsigned)(size_t)(buf + r * XPITCH + scol);
      asm volatile("global_load_async_to_lds_b128 %0, %1, %2"
                   :: "v"(laddr), "v"(goff), "s"(xbase) : "memory");
    }
  };

  const int wave  = t >> 5;
  const int lane  = t & 31;
  const int half  = lane >> 4;
  const int m16   = lane & 15;
  const int kpair = half << 1;
  const int wrow0 = row0 + (wave << 4);

  v8f c8 = {};

  auto compute = [&](int c) {
    const float* xa = &xs[c & 1][((wave << 4) + m16) * XPITCH + kpair];
    const float* wb = &lds_w[(c * CHUNK_K + kpair) * 16 + m16];
#pragma unroll 4
    for (int kk = 0; kk < CHUNK_K; kk += 4) {
      v2f a = *(const v2f*)(xa + kk);
      v2f b;
      b.x = wb[(kk + 0) * 16];
      b.y = wb[(kk + 1) * 16];
      c8 = __builtin_amdgcn_wmma_f32_16x16x4_f32(false, a, false, b,
                                                 (short)0, c8, false, false);
    }
  };

  // 2-deep async pipeline over 4 K-chunks
  issue(0);
  issue(1);
  WAIT_ASYNC(8);        // chunk 0 complete (async ops retire in order)
  __syncthreads();      // all waves' chunk-0 copies + lds_w visible
  compute(0);
  __syncthreads();      // everyone done reading buf0
  issue(2);
  WAIT_ASYNC(8);        // chunk 1 complete
  __syncthreads();
  compute(1);
  __syncthreads();
  issue(3);
  WAIT_ASYNC(8);        // chunk 2 complete
  __syncthreads();
  compute(2);
  __syncthreads();
  WAIT_ASYNC(0);        // chunk 3 complete
  __syncthreads();
  compute(3);

  // Epilogue: fast path for full tiles (781 of 782 blocks)
  if (m16 < NCLASS) {
    if (wrow0 + 16 <= N_NODES) {
      float* hp = hc + (size_t)(wrow0 + (half << 3)) * NCLASS + m16;
#pragma unroll
      for (int j = 0; j < 8; ++j) hp[(size_t)j * NCLASS] = c8[j];
    } else {
#pragma unroll
      for (int j = 0; j < 8; ++j) {
        int row = wrow0 + j + (half << 3);
        if (row < N_NODES) hc[(size_t)row * NCLASS + m16] = c8[j];
      }
    }
  }
}

// ---------------------------------------------------------------------------
// out[i] = dinv[i]^2 * hc[i] + bc   (self-loop term + bias, initializes d_out)
// ---------------------------------------------------------------------------
__global__ __launch_bounds__(256) void self_init_kernel(
    const float* __restrict__ hc, const float* __restrict__ dinv,
    const float* __restrict__ bc, float* __restrict__ out) {
  int i = blockIdx.x * blockDim.x + threadIdx.x;
  if (i >= N_NODES) return;
  float di = dinv[i];
  float w  = di * di;
  const float4* h4 = (const float4*)(hc + (size_t)i * NCLASS);
  float4 h0 = h4[0], h1 = h4[1];
  const float4* b4 = (const float4*)bc;
  float4 b0 = b4[0], b1 = b4[1];
  float4 o0, o1;
  o0.x = w * h0.x + b0.x;  o0.y = w * h0.y + b0.y;
  o0.z = w * h0.z + b0.z;  o0.w = w * h0.w + b0.w;
  o1.x = w * h1.x + b1.x;  o1.y = w * h1.y + b1.y;
  o1.z = w * h1.z + b1.z;  o1.w = w * h1.w + b1.w;
  float4* o4 = (float4*)(out + (size_t)i * NCLASS);
  o4[0] = o0; o4[1] = o1;
}

// ---------------------------------------------------------------------------
// Edge scatter: out[dst] += dinv[src]*dinv[dst] * hc[src]  (8 f32 HW atomics)
// hc/out/dinv all fit in the 192MB L2 -> L2-rate atomics.
// ---------------------------------------------------------------------------
__global__ __launch_bounds__(256) void scatter_kernel(
    const int* __restrict__ src, const int* __restrict__ dst,
    const float* __restrict__ dinv, const float* __restrict__ hc,
    float* __restrict__ out) {
  int e = blockIdx.x * blockDim.x + threadIdx.x;
  if (e >= N_EDGES) return;
  int s = src[e], d = dst[e];
  float w = dinv[s] * dinv[d];
  const float4* h4 = (const float4*)(hc + (size_t)s * NCLASS);
  float4 h0 = h4[0], h1 = h4[1];
  float* o = out + (size_t)d * NCLASS;
  unsafeAtomicAdd(o + 0, w * h0.x);
  unsafeAtomicAdd(o + 1, w * h0.y);
  unsafeAtomicAdd(o + 2, w * h0.z);
  unsafeAtomicAdd(o + 3, w * h0.w);
  unsafeAtomicAdd(o + 4, w * h1.x);
  unsafeAtomicAdd(o + 5, w * h1.y);
  unsafeAtomicAdd(o + 6, w * h1.z);
  unsafeAtomicAdd(o + 7, w * h1.w);
}

// ---------------------------------------------------------------------------
extern "C" void kernel_launch(void* const* d_in, const int* in_sizes, int n_in,
                              void* d_out, int out_size, void* d_ws, size_t ws_size,
                              hipStream_t stream) {
  const float* x   = (const float*)d_in[0];
  const int*   ei  = (const int*)d_in[1];   // [2, E] flat: src then dst
  const float* W1  = (const float*)d_in[2];
  const float* b1  = (const float*)d_in[3];
  const float* Wfc = (const float*)d_in[4];
  const float* bfc = (const float*)d_in[5];
  float* out = (float*)d_out;

  float* ws  = (float*)d_ws;
  float* deg = ws;               // 100000 floats; becomes dinv in place
  float* Wc  = ws + 100352;      // 2048 floats (256x8)
  float* bc  = ws + 102400;      // 8 floats (16B aligned)
  float* hc  = ws + 102416;      // 800000 floats (32B aligned)

  const int* srcp = ei;
  const int* dstp = ei + N_EDGES;

  precompute_kernel<<<1, 256, 0, stream>>>(W1, b1, Wfc, bfc, Wc, bc);
  deg_init_kernel<<<(N_NODES + 255) / 256, 256, 0, stream>>>(deg);
  deg_edge_kernel<<<(N_EDGES + 255) / 256, 256, 0, stream>>>(dstp, deg);
  dinv_kernel<<<(N_NODES + 255) / 256, 256, 0, stream>>>(deg);
  gemm_wmma_kernel<<<(N_NODES + TILE_ROWS - 1) / TILE_ROWS, 256, 0, stream>>>(x, Wc, hc);
  self_init_kernel<<<(N_NODES + 255) / 256, 256, 0, stream>>>(hc, deg, bc, out);
  scatter_kernel<<<(N_EDGES + 255) / 256, 256, 0, stream>>>(srcp, dstp, deg, hc, out);

  (void)in_sizes; (void)n_in; (void)out_size; (void)ws_size;
}